// Attention_19112604467516
// MI455X (gfx1250) — compile-verified
//
#include <hip/hip_runtime.h>

// Problem constants (match reference)
#define B_      32
#define S_      8192
#define D_      256
#define SC_     512                 // seq chunk per pass-1 block
#define NCHUNK_ (S_ / SC_)          // 16
#define T1_     256                 // pass-1 block threads (8 waves of 32)
#define NW_     (T1_ / 32)
#define NEG_SLOPE_ 0.2f
#define MASK_VAL_  (-10000.0f)

typedef float v2f __attribute__((ext_vector_type(2)));
typedef float v8f __attribute__((ext_vector_type(8)));

// Workspace layout (float offsets)
#define WS_M   0                        // [B_*NCHUNK_]  chunk max
#define WS_L   (WS_M + B_*NCHUNK_)      // [B_*NCHUNK_]  chunk sumexp
#define WS_U   (WS_L + B_*NCHUNK_)      // [B_*NCHUNK_*D_] chunk weighted-hs
#define WS_MB  (WS_U + B_*NCHUNK_*D_)   // [B_] global max
#define WS_ZB  (WS_MB + B_)             // [B_] global Z
#define WS_UB  (WS_ZB + B_)             // [B_*D_] global weighted-hs
#define WS_V   (WS_UB + B_*D_)          // [B_*D_] v = ht @ Wa_w

// ---------------------------------------------------------------------------
// Kernel 1: v[b,e] = sum_d ht[b,d] * Wa_w[d,e]    (M=32, N=256, K=256, f32)
// One wave per 16x16 tile, V_WMMA_F32_16X16X4_F32, K-loop of 64.
// A layout (32b, 16x4): lanes 0-15 = rows M, {v0,v1}={K,K+1}; lanes 16-31 same
// rows, {K+2,K+3}.  B layout symmetric: lanes 0-15 cols N with {K,K+1}, lanes
// 16-31 {K+2,K+3}.  C/D: VGPR r = rows {r, r+8} across lane halves.
// ---------------------------------------------------------------------------
__global__ __launch_bounds__(32)
void k_gemm_v(const float* __restrict__ ht, const float* __restrict__ Wa,
              float* __restrict__ v) {
  const int mt = blockIdx.x >> 4;        // 0..1
  const int nt = blockIdx.x & 15;        // 0..15
  const int lane = threadIdx.x;
  const int half = lane >> 4;
  const int l16  = lane & 15;
  v8f acc = {0.f, 0.f, 0.f, 0.f, 0.f, 0.f, 0.f, 0.f};
  const float* arow = ht + (mt * 16 + l16) * D_;
  for (int k = 0; k < D_; k += 4) {
    v2f a, b;
    a.x = arow[k + half * 2];
    a.y = arow[k + half * 2 + 1];
    b.x = Wa[(k + half * 2)     * D_ + nt * 16 + l16];
    b.y = Wa[(k + half * 2 + 1) * D_ + nt * 16 + l16];
    acc = __builtin_amdgcn_wmma_f32_16x16x4_f32(false, a, false, b,
                                                (short)0, acc, false, false);
  }
#pragma unroll
  for (int r = 0; r < 8; ++r)
    v[(mt * 16 + r + half * 8) * D_ + nt * 16 + l16] = acc[r];
}

// ---------------------------------------------------------------------------
// Kernel 2: fused streaming pass over hs (single HBM read of 268 MB).
// energy -> leaky_relu -> mask -> store raw energy into alpha buffer,
// online softmax accumulation of (m, l, u = sum w*hs) per wave, merged
// across the 8 waves through LDS; per-chunk partials to workspace.
// ---------------------------------------------------------------------------
__global__ __launch_bounds__(T1_)
void k_pass1(const float* __restrict__ hs, const float* __restrict__ ht,
             const float* __restrict__ Wa_b, const int* __restrict__ slen_p,
             const float* __restrict__ v, float* __restrict__ energy,
             float* __restrict__ ws) {
  const int ch = blockIdx.x;             // chunk
  const int b  = blockIdx.y;             // batch
  const int tid = threadIdx.x, lane = tid & 31, wv = tid >> 5;

  __shared__ float red[T1_];
  __shared__ float sm[NW_], sl[NW_];
  __shared__ float su[NW_][D_];          // 8 KB

  // c0 = ht[b] . Wa_b (cheap per-block recompute)
  red[tid] = ht[b * D_ + tid] * Wa_b[tid];
  __syncthreads();
  for (int st = T1_ / 2; st > 0; st >>= 1) {
    if (tid < st) red[tid] += red[tid + st];
    __syncthreads();
  }
  const float c0 = red[0];

  // each lane owns k-slice [lane*8, lane*8+8) of v / hs rows
  const float4* vv = (const float4*)(v + b * D_);
  const float4 va = vv[lane * 2 + 0];
  const float4 vb = vv[lane * 2 + 1];
  const int slen = slen_p[b];

  float m = -3.402823466e38f, l = 0.f;
  float u[8];
#pragma unroll
  for (int j = 0; j < 8; ++j) u[j] = 0.f;

  for (int s0 = wv; s0 < SC_; s0 += NW_) {   // 64 rows per wave
    const int sg = ch * SC_ + s0;
    const float4* hp = (const float4*)(hs + ((size_t)b * S_ + sg) * D_);
    const float4 h0 = hp[lane * 2 + 0];      // wave reads 1 KB contiguous
    const float4 h1 = hp[lane * 2 + 1];
    float dp = h0.x * va.x + h0.y * va.y + h0.z * va.z + h0.w * va.w
             + h1.x * vb.x + h1.y * vb.y + h1.z * vb.z + h1.w * vb.w;
#pragma unroll
    for (int off = 16; off >= 1; off >>= 1) dp += __shfl_xor(dp, off, 32);
    float e = dp + c0;
    e = (e > 0.f) ? e : NEG_SLOPE_ * e;
    if (sg >= slen) e = MASK_VAL_;
    if (lane == 0) energy[(size_t)b * S_ + sg] = e;  // raw masked energy

    float w;
    if (e > m) {                     // online-softmax rescale
      const float sc = __expf(m - e);
      l *= sc;
#pragma unroll
      for (int j = 0; j < 8; ++j) u[j] *= sc;
      m = e; w = 1.f;
    } else {
      w = __expf(e - m);
    }
    l += w;
    u[0] += w * h0.x; u[1] += w * h0.y; u[2] += w * h0.z; u[3] += w * h0.w;
    u[4] += w * h1.x; u[5] += w * h1.y; u[6] += w * h1.z; u[7] += w * h1.w;
  }

  if (lane == 0) { sm[wv] = m; sl[wv] = l; }
#pragma unroll
  for (int j = 0; j < 8; ++j) su[wv][lane * 8 + j] = u[j];
  __syncthreads();

  float mb = -3.402823466e38f;
  for (int w2 = 0; w2 < NW_; ++w2) mb = fmaxf(mb, sm[w2]);
  float ub = 0.f;
  for (int w2 = 0; w2 < NW_; ++w2) ub += __expf(sm[w2] - mb) * su[w2][tid];
  ws[WS_U + ((size_t)b * NCHUNK_ + ch) * D_ + tid] = ub;
  if (tid == 0) {
    float lb = 0.f;
    for (int w2 = 0; w2 < NW_; ++w2) lb += sl[w2] * __expf(sm[w2] - mb);
    ws[WS_M + b * NCHUNK_ + ch] = mb;
    ws[WS_L + b * NCHUNK_ + ch] = lb;
  }
}

// ---------------------------------------------------------------------------
// Kernel 3: combine chunk partials per batch -> m_b, Z_b, u_b
// ---------------------------------------------------------------------------
__global__ __launch_bounds__(D_)
void k_combine(float* __restrict__ ws) {
  const int b = blockIdx.x, tid = threadIdx.x;
  float mb = -3.402823466e38f;
  for (int c = 0; c < NCHUNK_; ++c)
    mb = fmaxf(mb, ws[WS_M + b * NCHUNK_ + c]);
  float zb = 0.f, ub = 0.f;
  for (int c = 0; c < NCHUNK_; ++c) {
    const float sc = __expf(ws[WS_M + b * NCHUNK_ + c] - mb);
    zb += sc * ws[WS_L + b * NCHUNK_ + c];
    ub += sc * ws[WS_U + ((size_t)b * NCHUNK_ + c) * D_ + tid];
  }
  ws[WS_UB + b * D_ + tid] = ub;
  if (tid == 0) { ws[WS_MB + b] = mb; ws[WS_ZB + b] = zb; }
}

// ---------------------------------------------------------------------------
// Kernel 4: normalize alpha in place: alpha = exp(e - m_b) / Z_b
// ---------------------------------------------------------------------------
__global__ __launch_bounds__(256)
void k_alpha(float* __restrict__ alpha, const float* __restrict__ ws) {
  const int idx = blockIdx.x * 256 + threadIdx.x;
  const int b = idx >> 13;                       // S_ = 8192
  const float inv = 1.f / ws[WS_ZB + b];
  alpha[idx] = __expf(alpha[idx] - ws[WS_MB + b]) * inv;
}

// ---------------------------------------------------------------------------
// Kernel 5: context[b,c] = sum_e u_b[b,e] * Wc_w[c,e] + Wc_b[c]   (WMMA)
// B[k=e][n=c] = Wc_w[c*D_ + e]
// ---------------------------------------------------------------------------
__global__ __launch_bounds__(32)
void k_gemm_ctx(const float* __restrict__ ws, const float* __restrict__ Wc,
                const float* __restrict__ Wc_b, float* __restrict__ out) {
  const float* U = ws + WS_UB;
  const int mt = blockIdx.x >> 4;
  const int nt = blockIdx.x & 15;
  const int lane = threadIdx.x;
  const int half = lane >> 4;
  const int l16  = lane & 15;
  v8f acc = {0.f, 0.f, 0.f, 0.f, 0.f, 0.f, 0.f, 0.f};
  const float* arow = U  + (mt * 16 + l16) * D_;
  const float* bcol = Wc + (nt * 16 + l16) * D_;   // row c of Wc_w = column n of B
  for (int k = 0; k < D_; k += 4) {
    v2f a, b;
    a.x = arow[k + half * 2];
    a.y = arow[k + half * 2 + 1];
    b.x = bcol[k + half * 2];
    b.y = bcol[k + half * 2 + 1];
    acc = __builtin_amdgcn_wmma_f32_16x16x4_f32(false, a, false, b,
                                                (short)0, acc, false, false);
  }
  const float bias = Wc_b[nt * 16 + l16];
#pragma unroll
  for (int r = 0; r < 8; ++r)
    out[(mt * 16 + r + half * 8) * D_ + nt * 16 + l16] = acc[r] + bias;
}

// ---------------------------------------------------------------------------
extern "C" void kernel_launch(void* const* d_in, const int* in_sizes, int n_in,
                              void* d_out, int out_size, void* d_ws, size_t ws_size,
                              hipStream_t stream) {
  (void)in_sizes; (void)n_in; (void)out_size; (void)ws_size;
  const float* hs    = (const float*)d_in[0];
  const float* ht    = (const float*)d_in[1];
  const int*   slen  = (const int*)  d_in[2];
  const float* Wa_w  = (const float*)d_in[3];
  const float* Wa_b  = (const float*)d_in[4];
  const float* Wc_w  = (const float*)d_in[5];
  const float* Wc_b  = (const float*)d_in[6];

  float* out     = (float*)d_out;
  float* alpha   = out;               // [B_*S_]
  float* context = out + B_ * S_;     // [B_*D_]
  float* ws      = (float*)d_ws;
  float* v       = ws + WS_V;

  k_gemm_v  <<<32, 32, 0, stream>>>(ht, Wa_w, v);
  k_pass1   <<<dim3(NCHUNK_, B_), T1_, 0, stream>>>(hs, ht, Wa_b, slen, v, alpha, ws);
  k_combine <<<B_, D_, 0, stream>>>(ws);
  k_alpha   <<<(B_ * S_) / 256, 256, 0, stream>>>(alpha, ws);
  k_gemm_ctx<<<32, 32, 0, stream>>>(ws, Wc_w, Wc_b, context);
}